// GINConvNet2_44805098832501
// MI455X (gfx1250) — compile-verified
//
#include <hip/hip_runtime.h>
#include <hip/hip_bf16.h>

#define BN_EPS 1e-5f

typedef float v2f __attribute__((ext_vector_type(2)));
typedef float v8f __attribute__((ext_vector_type(8)));

// D = A(16x4 f32) * B(4x16 f32) + C(16x16 f32), full-precision f32 WMMA.
__device__ __forceinline__ v8f wmma4(v2f a, v2f b, v8f c) {
    return __builtin_amdgcn_wmma_f32_16x16x4_f32(
        /*neg_a=*/false, a, /*neg_b=*/false, b,
        /*c_mod=*/(short)0, c, /*reuse_a=*/false, /*reuse_b=*/false);
}

// Pack a KxNC row-major weight into fragment-ready pairs:
//   Wp[(k>>1)*NC*2 + n*2 + (k&1)] = W[k*NC + n]
// so the B fragment {W[kk][n], W[kk+1][n]} is one aligned 8-byte LDS load.
template <int K, int NC>
__device__ __forceinline__ void pack_pairs(const float* __restrict__ W,
                                           float* __restrict__ Wp) {
    for (int i = threadIdx.x; i < K * NC; i += 256) {
        int k = i / NC, n = i % NC;
        Wp[(k >> 1) * (NC * 2) + n * 2 + (k & 1)] = W[i];
    }
}

#define ASTRIDE 34  // even (8B-aligned v2f) + conflict-free for lanes 0..15

// ---------------------------------------------------------------------------
// y[N,32] = x[N,128] @ W[128,32]   (layer-1 projection pushed ahead of agg)
// ---------------------------------------------------------------------------
__global__ __launch_bounds__(256) void k_proj128(const float* __restrict__ x,
                                                 const float* __restrict__ W,
                                                 float* __restrict__ y, int N) {
    __shared__ float Wp[128 * 32];
    pack_pairs<128, 32>(W, Wp);
    __syncthreads();

    const int wave = threadIdx.x >> 5, lane = threadIdx.x & 31;
    const int half = lane >> 4, l16 = lane & 15;
    const int tile = blockIdx.x * 8 + wave;
    if (tile >= (N + 15) >> 4) return;

    int row = tile * 16 + l16;
    int rowc = row < N ? row : N - 1;
    const float* xr = x + (size_t)rowc * 128;

    v8f c0 = {}, c1 = {};
    for (int k0 = 0; k0 < 128; k0 += 4) {
        const int kk = k0 + 2 * half;
        v2f a = *(const v2f*)(xr + kk);
        const float* wrow = Wp + (kk >> 1) * 64;
        v2f b0 = *(const v2f*)(wrow + l16 * 2);
        v2f b1 = *(const v2f*)(wrow + (16 + l16) * 2);
        c0 = wmma4(a, b0, c0);
        c1 = wmma4(a, b1, c1);
    }
    const int base = tile * 16 + 8 * half;
    for (int r = 0; r < 8; ++r) {
        int orow = base + r;
        if (orow < N) {
            y[(size_t)orow * 32 + l16] = c0[r];
            y[(size_t)orow * 32 + 16 + l16] = c1[r];
        }
    }
}

// ---------------------------------------------------------------------------
__global__ __launch_bounds__(256) void k_zero(float4* __restrict__ p, int n4) {
    int i = blockIdx.x * blockDim.x + threadIdx.x;
    if (i < n4) p[i] = make_float4(0.f, 0.f, 0.f, 0.f);
}

// agg[dst] += h[src]; one edge per wave, lane = feature (128B coalesced both sides).
__global__ __launch_bounds__(256) void k_edge_agg(const int* __restrict__ ei,
                                                  const float* __restrict__ h,
                                                  float* __restrict__ agg, int E) {
    int t = blockIdx.x * blockDim.x + threadIdx.x;
    int e = t >> 5, lane = t & 31;
    if (e >= E) return;
    int s = ei[e];
    int d = ei[E + e];
    atomicAdd(&agg[(size_t)d * 32 + lane], h[(size_t)s * 32 + lane]);
}

// 16x32 @ 32x32 GEMM core: A from wave-private LDS (stride ASTRIDE),
// Wp packed-pair LDS. Produces both 16-wide column halves.
__device__ __forceinline__ void gemm32(const float* __restrict__ A,
                                       const float* __restrict__ Wp,
                                       int half, int l16, v8f& c0, v8f& c1) {
    for (int k0 = 0; k0 < 32; k0 += 4) {
        const int kk = k0 + 2 * half;
        v2f a = *(const v2f*)(A + l16 * ASTRIDE + kk);
        const float* wrow = Wp + (kk >> 1) * 64;
        v2f b0 = *(const v2f*)(wrow + l16 * 2);
        v2f b1 = *(const v2f*)(wrow + (16 + l16) * 2);
        c0 = wmma4(a, b0, c0);
        c1 = wmma4(a, b1, c1);
    }
}

// ---------------------------------------------------------------------------
// Layer-1 tail: t = relu(y + agg + b1a); z = BN(relu(t @ W1b + b1b))
// ---------------------------------------------------------------------------
__global__ __launch_bounds__(256) void k_gin1_tail(
    const float* __restrict__ y, const float* __restrict__ agg,
    const float* __restrict__ ba, const float* __restrict__ Wb,
    const float* __restrict__ bb, const float* __restrict__ g,
    const float* __restrict__ be, const float* __restrict__ m,
    const float* __restrict__ v, float* __restrict__ z, int N) {
    __shared__ float Wp[32 * 32];
    __shared__ float As[8][16 * ASTRIDE];
    pack_pairs<32, 32>(Wb, Wp);
    __syncthreads();

    const int wave = threadIdx.x >> 5, lane = threadIdx.x & 31;
    const int half = lane >> 4, l16 = lane & 15;
    const int tile = blockIdx.x * 8 + wave;
    if (tile >= (N + 15) >> 4) return;
    float* A = As[wave];

    {   // A = relu(y + agg + ba)
        int rrow = lane & 15;
        int grow = tile * 16 + rrow; grow = grow < N ? grow : N - 1;
        int cbase = (lane >> 4) * 16;
        const float4* yp = (const float4*)(y + (size_t)grow * 32 + cbase);
        const float4* ap = (const float4*)(agg + (size_t)grow * 32 + cbase);
        const float4* bp = (const float4*)(ba + cbase);
        for (int q = 0; q < 4; ++q) {
            float4 yv = yp[q], av = ap[q], bv = bp[q];
            float* dst = &A[rrow * ASTRIDE + cbase + q * 4];
            dst[0] = fmaxf(yv.x + av.x + bv.x, 0.f);
            dst[1] = fmaxf(yv.y + av.y + bv.y, 0.f);
            dst[2] = fmaxf(yv.z + av.z + bv.z, 0.f);
            dst[3] = fmaxf(yv.w + av.w + bv.w, 0.f);
        }
    }

    v8f c0 = {}, c1 = {};
    gemm32(A, Wp, half, l16, c0, c1);

    for (int j = 0; j < 2; ++j) {
        int col = j * 16 + l16;
        float bbv = bb[col];
        float s = g[col] * rsqrtf(v[col] + BN_EPS);
        float o = be[col] - m[col] * s;
        v8f c = j ? c1 : c0;
        int base = tile * 16 + 8 * half;
        for (int r = 0; r < 8; ++r) {
            int orow = base + r;
            if (orow < N)
                z[(size_t)orow * 32 + col] = fmaxf(c[r] + bbv, 0.f) * s + o;
        }
    }
}

// ---------------------------------------------------------------------------
// Full GIN layer (layers 2,3): h = z+agg; u = relu(h@Wa+ba);
// zout = BN(relu(u@Wb+bb)). Two chained WMMA GEMMs via wave-private LDS.
// ---------------------------------------------------------------------------
__global__ __launch_bounds__(256) void k_gin_layer(
    const float* __restrict__ zin, const float* __restrict__ agg,
    const float* __restrict__ Wa, const float* __restrict__ ba,
    const float* __restrict__ Wb, const float* __restrict__ bb,
    const float* __restrict__ g, const float* __restrict__ be,
    const float* __restrict__ m, const float* __restrict__ v,
    float* __restrict__ zout, int N) {
    __shared__ float Wap[32 * 32];
    __shared__ float Wbp[32 * 32];
    __shared__ float As[8][16 * ASTRIDE];
    pack_pairs<32, 32>(Wa, Wap);
    pack_pairs<32, 32>(Wb, Wbp);
    __syncthreads();

    const int wave = threadIdx.x >> 5, lane = threadIdx.x & 31;
    const int half = lane >> 4, l16 = lane & 15;
    const int tile = blockIdx.x * 8 + wave;
    if (tile >= (N + 15) >> 4) return;
    float* A = As[wave];

    {   // A = zin + agg
        int rrow = lane & 15;
        int grow = tile * 16 + rrow; grow = grow < N ? grow : N - 1;
        int cbase = (lane >> 4) * 16;
        const float4* zp = (const float4*)(zin + (size_t)grow * 32 + cbase);
        const float4* ap = (const float4*)(agg + (size_t)grow * 32 + cbase);
        for (int q = 0; q < 4; ++q) {
            float4 zv = zp[q], av = ap[q];
            float* dst = &A[rrow * ASTRIDE + cbase + q * 4];
            dst[0] = zv.x + av.x; dst[1] = zv.y + av.y;
            dst[2] = zv.z + av.z; dst[3] = zv.w + av.w;
        }
    }

    // GEMM1: u = relu(A @ Wa + ba) -> back into A
    v8f c0 = {}, c1 = {};
    gemm32(A, Wap, half, l16, c0, c1);
    for (int j = 0; j < 2; ++j) {
        int col = j * 16 + l16;
        float bav = ba[col];
        v8f c = j ? c1 : c0;
        for (int r = 0; r < 8; ++r)
            A[(r + 8 * half) * ASTRIDE + col] = fmaxf(c[r] + bav, 0.f);
    }

    // GEMM2: z = BN(relu(u @ Wb + bb))
    v8f d0 = {}, d1 = {};
    gemm32(A, Wbp, half, l16, d0, d1);
    for (int j = 0; j < 2; ++j) {
        int col = j * 16 + l16;
        float bbv = bb[col];
        float s = g[col] * rsqrtf(v[col] + BN_EPS);
        float o = be[col] - m[col] * s;
        v8f d = j ? d1 : d0;
        int base = tile * 16 + 8 * half;
        for (int r = 0; r < 8; ++r) {
            int orow = base + r;
            if (orow < N)
                zout[(size_t)orow * 32 + col] = fmaxf(d[r] + bbv, 0.f) * s + o;
        }
    }
}

// ---------------------------------------------------------------------------
// Head: u = relu(z@Wf1+bf1); logits = u@Wf2+bf2 (Wf2 zero-padded 32x16);
// out = log_softmax(logits[:, :10])
// ---------------------------------------------------------------------------
__global__ __launch_bounds__(256) void k_head(const float* __restrict__ z,
                                              const float* __restrict__ Wf1,
                                              const float* __restrict__ bf1,
                                              const float* __restrict__ Wf2,
                                              const float* __restrict__ bf2,
                                              float* __restrict__ out, int N) {
    __shared__ float W1p[32 * 32];
    __shared__ float W2p[32 * 16];
    __shared__ float As[8][16 * ASTRIDE];
    pack_pairs<32, 32>(Wf1, W1p);
    for (int i = threadIdx.x; i < 32 * 16; i += 256) {
        int k = i >> 4, n = i & 15;
        W2p[(k >> 1) * 32 + n * 2 + (k & 1)] = (n < 10) ? Wf2[k * 10 + n] : 0.f;
    }
    __syncthreads();

    const int wave = threadIdx.x >> 5, lane = threadIdx.x & 31;
    const int half = lane >> 4, l16 = lane & 15;
    const int tile = blockIdx.x * 8 + wave;
    if (tile >= (N + 15) >> 4) return;
    float* A = As[wave];

    {   // A = z tile
        int rrow = lane & 15;
        int grow = tile * 16 + rrow; grow = grow < N ? grow : N - 1;
        int cbase = (lane >> 4) * 16;
        const float4* zp = (const float4*)(z + (size_t)grow * 32 + cbase);
        for (int q = 0; q < 4; ++q) {
            float4 zv = zp[q];
            float* dst = &A[rrow * ASTRIDE + cbase + q * 4];
            dst[0] = zv.x; dst[1] = zv.y; dst[2] = zv.z; dst[3] = zv.w;
        }
    }

    // GEMM1: u = relu(A @ Wf1 + bf1) -> back into A
    v8f c0 = {}, c1 = {};
    gemm32(A, W1p, half, l16, c0, c1);
    for (int j = 0; j < 2; ++j) {
        int col = j * 16 + l16;
        float bv = bf1[col];
        v8f c = j ? c1 : c0;
        for (int r = 0; r < 8; ++r)
            A[(r + 8 * half) * ASTRIDE + col] = fmaxf(c[r] + bv, 0.f);
    }

    // GEMM2: logits (16 cols, 10 valid) -> stash into A[:, 0..15]
    v8f d = {};
    for (int k0 = 0; k0 < 32; k0 += 4) {
        const int kk = k0 + 2 * half;
        v2f a = *(const v2f*)(A + l16 * ASTRIDE + kk);
        v2f b0 = *(const v2f*)(W2p + (kk >> 1) * 32 + l16 * 2);
        d = wmma4(a, b0, d);
    }
    float bf2v = (l16 < 10) ? bf2[l16] : 0.f;
    for (int r = 0; r < 8; ++r)
        A[(r + 8 * half) * ASTRIDE + l16] = d[r] + bf2v;

    // log_softmax: lanes 0..15 each own one node row
    if (half == 0) {
        int grow = tile * 16 + l16;
        if (grow < N) {
            const float* Lg = &A[l16 * ASTRIDE];
            float mx = Lg[0];
            for (int c = 1; c < 10; ++c) mx = fmaxf(mx, Lg[c]);
            float sum = 0.f;
            for (int c = 0; c < 10; ++c) sum += expf(Lg[c] - mx);
            float lse = logf(sum) + mx;
            for (int c = 0; c < 10; ++c)
                out[(size_t)grow * 10 + c] = Lg[c] - lse;
        }
    }
}

// ---------------------------------------------------------------------------
extern "C" void kernel_launch(void* const* d_in, const int* in_sizes, int n_in,
                              void* d_out, int out_size, void* d_ws, size_t ws_size,
                              hipStream_t stream) {
    const float* x   = (const float*)d_in[0];
    const int*   ei  = (const int*)d_in[1];
    const float* W1a = (const float*)d_in[2];  const float* b1a = (const float*)d_in[3];
    const float* W1b = (const float*)d_in[4];  const float* b1b = (const float*)d_in[5];
    const float* W2a = (const float*)d_in[6];  const float* b2a = (const float*)d_in[7];
    const float* W2b = (const float*)d_in[8];  const float* b2b = (const float*)d_in[9];
    const float* W3a = (const float*)d_in[10]; const float* b3a = (const float*)d_in[11];
    const float* W3b = (const float*)d_in[12]; const float* b3b = (const float*)d_in[13];
    const float* g1 = (const float*)d_in[14];  const float* be1 = (const float*)d_in[15];
    const float* m1 = (const float*)d_in[16];  const float* v1  = (const float*)d_in[17];
    const float* g2 = (const float*)d_in[18];  const float* be2 = (const float*)d_in[19];
    const float* m2 = (const float*)d_in[20];  const float* v2  = (const float*)d_in[21];
    const float* g3 = (const float*)d_in[22];  const float* be3 = (const float*)d_in[23];
    const float* m3 = (const float*)d_in[24];  const float* v3  = (const float*)d_in[25];
    const float* Wf1 = (const float*)d_in[26]; const float* bf1 = (const float*)d_in[27];
    const float* Wf2 = (const float*)d_in[28]; const float* bf2 = (const float*)d_in[29];

    const int N = in_sizes[0] / 128;
    const int E = in_sizes[1] / 2;
    float* out = (float*)d_out;

    float* bufA = (float*)d_ws;             // N x 32
    float* bufB = bufA + (size_t)N * 32;    // N x 32 (agg)
    float* bufC = bufB + (size_t)N * 32;    // N x 32

    const int nTiles = (N + 15) / 16;
    dim3 b256(256);
    dim3 gNode((nTiles + 7) / 8);
    const int n4 = (N * 32) / 4;
    dim3 gZero((n4 + 255) / 256);
    dim3 gEdge((unsigned)(((long long)E * 32 + 255) / 256));

    // L1: project first (agg commutes with the linear), then aggregate 32-wide
    k_proj128<<<gNode, b256, 0, stream>>>(x, W1a, bufA, N);
    k_zero<<<gZero, b256, 0, stream>>>((float4*)bufB, n4);
    k_edge_agg<<<gEdge, b256, 0, stream>>>(ei, bufA, bufB, E);
    k_gin1_tail<<<gNode, b256, 0, stream>>>(bufA, bufB, b1a, W1b, b1b,
                                            g1, be1, m1, v1, bufC, N);
    // L2
    k_zero<<<gZero, b256, 0, stream>>>((float4*)bufB, n4);
    k_edge_agg<<<gEdge, b256, 0, stream>>>(ei, bufC, bufB, E);
    k_gin_layer<<<gNode, b256, 0, stream>>>(bufC, bufB, W2a, b2a, W2b, b2b,
                                            g2, be2, m2, v2, bufA, N);
    // L3
    k_zero<<<gZero, b256, 0, stream>>>((float4*)bufB, n4);
    k_edge_agg<<<gEdge, b256, 0, stream>>>(ei, bufA, bufB, E);
    k_gin_layer<<<gNode, b256, 0, stream>>>(bufA, bufB, W3a, b3a, W3b, b3b,
                                            g3, be3, m3, v3, bufC, N);
    // Head
    k_head<<<gNode, b256, 0, stream>>>(bufC, Wf1, bf1, Wf2, bf2, out, N);
}